// SwitchboardAttention_11364483465951
// MI455X (gfx1250) — compile-verified
//
#include <hip/hip_runtime.h>
#include <hip/hip_bf16.h>

// Problem constants (match reference)
#define BB 64
#define TT 4096
#define NN 512
#define LL 64            // chunk length along T
#define CC (TT / LL)     // 64 chunks
#define DSTR 68          // per-(b,c) chunk-polynomial stride in ws (needs 65)
#define CTSTR 72         // LDS stride for Ct rows (2*72 mod 64 = 16 -> no bank overlap)

typedef __attribute__((ext_vector_type(2))) float v2f;
typedef __attribute__((ext_vector_type(8))) float v8f;

__device__ __forceinline__ float sigmoidf(float v) {
    return 1.0f / (1.0f + __expf(-v));
}

// ---------------------------------------------------------------------------
// K1: per (b, chunk) build the chunk transfer polynomial
//     P(S) = prod_{t in chunk} ((1-g_t) + g_t S), store coeffs D[0..LL] to ws.
//     One wave per chunk; 2 coefficients per lane; neighbor via __shfl_up.
// ---------------------------------------------------------------------------
__global__ __launch_bounds__(32) void chunk_poly_kernel(const float* __restrict__ x,
                                                        float* __restrict__ Dws) {
    const int c = blockIdx.x;
    const int b = blockIdx.y;
    const int lane = threadIdx.x;  // 0..31, k-slots {2*lane, 2*lane+1}
    float p0 = (lane == 0) ? 1.0f : 0.0f;
    float p1 = 0.0f;
    float gprod = 1.0f;  // top coefficient S^LL = prod(g)
    const float* xrow = x + (size_t)b * TT + c * LL;
    for (int tau = 0; tau < LL; ++tau) {
        float gt = sigmoidf(xrow[tau]);
        gprod *= gt;
        float pm1 = __shfl_up(p1, 1);   // p[2*lane - 1] (old value)
        if (lane == 0) pm1 = 0.0f;
        float om = 1.0f - gt;
        p1 = om * p1 + gt * p0;
        p0 = om * p0 + gt * pm1;
    }
    float* D = Dws + ((size_t)b * CC + c) * DSTR;
    D[2 * lane + 0] = p0;
    D[2 * lane + 1] = p1;
    if (lane == 0) D[LL] = gprod;
}

// ---------------------------------------------------------------------------
// K2: per batch, serial scan over chunk boundaries:
//     bnd_{c+1}[j] = sum_k D_c[k] * bnd_c[j-k]   (short convolution)
//     Stores every boundary state to ws for K3.
// ---------------------------------------------------------------------------
__global__ __launch_bounds__(NN) void boundary_scan_kernel(const float* __restrict__ Dws,
                                                           float* __restrict__ Bnd) {
    const int b = blockIdx.x;
    const int j = threadIdx.x;  // 0..511
    __shared__ float sb[LL + NN];   // front-padded with LL zeros
    __shared__ float sD[LL + 4];
    for (int i = j; i < LL + NN; i += NN) sb[i] = 0.0f;
    __syncthreads();
    if (j == 0) sb[LL] = 1.0f;      // initial state e_0
    __syncthreads();
    for (int c = 0; c < CC; ++c) {
        Bnd[((size_t)b * CC + c) * NN + j] = sb[LL + j];
        if (c == CC - 1) break;
        if (j <= LL) sD[j] = Dws[((size_t)b * CC + c) * DSTR + j];
        __syncthreads();
        float acc = 0.0f;
        #pragma unroll 4
        for (int k = 0; k <= LL; ++k)
            acc += sD[k] * sb[LL + j - k];
        __syncthreads();
        sb[LL + j] = acc;
        __syncthreads();
    }
}

// ---------------------------------------------------------------------------
// K3: per (b, chunk): Out(512 x 64) = A(512 x 64) * B(64 x 64) via f32 WMMA.
//     Reversed-k formulation so A-fragment loads ascend in LDS:
//       Out[j,tau] = sum_q  sb[j+1+q] * Crev[q][tau],
//       Crev[q][tau] = g_t * coeff_{LL-1-q}(P_tau)     (built in LDS by wave 0).
//     8 waves; wave w owns row-tiles {w, w+8, w+16, w+24}. For each row-tile
//     we keep all 4 column-tile accumulators live so each A-fragment feeds
//     4 independent WMMA chains (1.25 LDS loads per wmma, latency hidden).
// ---------------------------------------------------------------------------
__global__ __launch_bounds__(256) void out_tile_kernel(const float* __restrict__ x,
                                                       const float* __restrict__ Bnd,
                                                       float* __restrict__ out) {
    const int c = blockIdx.x;
    const int b = blockIdx.y;
    const int tid = threadIdx.x;
    const int wave = tid >> 5;
    const int lane = tid & 31;
    __shared__ float sCt[LL * CTSTR];  // sCt[(LL-1-k)*CTSTR + tau] = g_t * p_tau[k]
    __shared__ float sb[LL + NN];      // zero pad + boundary state

    if (wave != 0) {
        // waves 1..7: stage boundary state (with zero front-pad)
        const float* bn = Bnd + ((size_t)b * CC + c) * NN;
        for (int i = tid - 32; i < LL + NN; i += 224)
            sb[i] = (i < LL) ? 0.0f : bn[i - LL];
    } else {
        // wave 0: rebuild chunk-local scaled coefficient matrix into LDS,
        // rows stored k-reversed so GEMM B-fragment rows ascend.
        float p0 = (lane == 0) ? 1.0f : 0.0f;
        float p1 = 0.0f;
        const float* xrow = x + (size_t)b * TT + c * LL;
        const int r0 = (LL - 1) - 2 * lane;  // row for coeff k=2*lane
        const int r1 = r0 - 1;               // row for coeff k=2*lane+1
        for (int tau = 0; tau < LL; ++tau) {
            float gt = sigmoidf(xrow[tau]);
            sCt[r0 * CTSTR + tau] = gt * p0;
            sCt[r1 * CTSTR + tau] = gt * p1;
            float pm1 = __shfl_up(p1, 1);
            if (lane == 0) pm1 = 0.0f;
            float om = 1.0f - gt;
            p1 = om * p1 + gt * p0;
            p0 = om * p0 + gt * pm1;
        }
    }
    __syncthreads();

    const int m  = lane & 15;            // M for A / N for B / column within tile
    const int hi = (lane >> 4) << 1;     // K offset 0 or 2 (half-wave split)
    const int rowoff = (lane >> 4) << 3; // D-matrix: lanes 16-31 hold M+8

    for (int jtw = 0; jtw < 4; ++jtw) {
        const int j0 = (wave + 8 * jtw) * 16;       // output row-tile base
        v8f acc0 = {}, acc1 = {}, acc2 = {}, acc3 = {};
        const int abase = j0 + m + 1 + hi;          // ascending A-frag base
        #pragma unroll
        for (int q0 = 0; q0 < LL; q0 += 4) {
            v2f a;
            a.x = sb[abase + q0];                   // A[j0+m, q0+hi]
            a.y = sb[abase + q0 + 1];               // A[j0+m, q0+hi+1]
            const int bbase = (q0 + hi) * CTSTR + m;
#if defined(__has_builtin) && __has_builtin(__builtin_amdgcn_wmma_f32_16x16x4_f32)
            {
                v2f bv;
                bv.x = sCt[bbase + 0];
                bv.y = sCt[bbase + CTSTR];
                acc0 = __builtin_amdgcn_wmma_f32_16x16x4_f32(false, a, false, bv,
                                                             (short)0, acc0, false, false);
                bv.x = sCt[bbase + 16];
                bv.y = sCt[bbase + CTSTR + 16];
                acc1 = __builtin_amdgcn_wmma_f32_16x16x4_f32(false, a, false, bv,
                                                             (short)0, acc1, false, false);
                bv.x = sCt[bbase + 32];
                bv.y = sCt[bbase + CTSTR + 32];
                acc2 = __builtin_amdgcn_wmma_f32_16x16x4_f32(false, a, false, bv,
                                                             (short)0, acc2, false, false);
                bv.x = sCt[bbase + 48];
                bv.y = sCt[bbase + CTSTR + 48];
                acc3 = __builtin_amdgcn_wmma_f32_16x16x4_f32(false, a, false, bv,
                                                             (short)0, acc3, false, false);
            }
#else
            // scalar fallback (keeps compile alive; histogram will show wmma=0)
            #pragma unroll
            for (int r = 0; r < 8; ++r) {
                #pragma unroll
                for (int kk = 0; kk < 4; ++kk) {
                    const float av = sb[j0 + rowoff + r + 1 + q0 + kk];
                    acc0[r] += av * sCt[(q0 + kk) * CTSTR + m];
                    acc1[r] += av * sCt[(q0 + kk) * CTSTR + m + 16];
                    acc2[r] += av * sCt[(q0 + kk) * CTSTR + m + 32];
                    acc3[r] += av * sCt[(q0 + kk) * CTSTR + m + 48];
                }
            }
#endif
        }
        // store 4 x (16x16) fp32 D tiles: VGPR r -> row (rowoff+r), col (t0+m)
        const size_t base =
            ((size_t)b * NN + (size_t)(j0 + rowoff)) * TT + (size_t)c * LL + m;
        #pragma unroll
        for (int r = 0; r < 8; ++r) {
            out[base + (size_t)r * TT +  0] = acc0[r];
            out[base + (size_t)r * TT + 16] = acc1[r];
            out[base + (size_t)r * TT + 32] = acc2[r];
            out[base + (size_t)r * TT + 48] = acc3[r];
        }
    }
}

// ---------------------------------------------------------------------------
extern "C" void kernel_launch(void* const* d_in, const int* in_sizes, int n_in,
                              void* d_out, int out_size, void* d_ws, size_t ws_size,
                              hipStream_t stream) {
    (void)in_sizes; (void)n_in; (void)out_size; (void)ws_size;
    const float* x = (const float*)d_in[0];
    float* out = (float*)d_out;
    float* Dws = (float*)d_ws;                              // BB*CC*DSTR floats
    float* Bnd = Dws + (size_t)BB * CC * DSTR;              // BB*CC*NN floats

    dim3 gchunks(CC, BB);
    hipLaunchKernelGGL(chunk_poly_kernel,    gchunks,  dim3(32),  0, stream, x, Dws);
    hipLaunchKernelGGL(boundary_scan_kernel, dim3(BB), dim3(NN), 0, stream, Dws, Bnd);
    hipLaunchKernelGGL(out_tile_kernel,      gchunks,  dim3(256), 0, stream, x, Bnd, out);
}